// ConvolutionModule_18305150615818
// MI455X (gfx1250) — compile-verified
//
#include <hip/hip_runtime.h>

typedef __attribute__((ext_vector_type(2))) float v2f;
typedef __attribute__((ext_vector_type(8))) float v8f;

#define NN   50000   // nodes
#define NE   800000  // edges
#define DIM  128     // feature / hidden size
#define NG   64      // graphs in batch

#define WSTRIDE 132  // padded LDS row stride (dwords): %64 == 4 -> conflict-free b64 frags
#define SMEM_FLOATS (2 * DIM * WSTRIDE + DIM)
#define SMEM_BYTES  (SMEM_FLOATS * 4)   // 135,680 B < 320 KB/WGP

// ---------------------------------------------------------------- utilities

__global__ void zero_kernel(float* __restrict__ p, long n) {
  long i = (long)blockIdx.x * blockDim.x + threadIdx.x;
  if (i < n) p[i] = 0.0f;
}

// ------------------------------------------------- handcrafted column stats

__global__ void colsum_kernel(const float* __restrict__ x, float* __restrict__ colsum) {
  int f = threadIdx.x;  // blockDim.x == 128 == DIM
  float s = 0.0f;
  for (int r = blockIdx.x; r < NN; r += gridDim.x)
    s += x[(long)r * DIM + f];
  unsafeAtomicAdd(&colsum[f], s);
}

__global__ void handcrafted_kernel(const float* __restrict__ colsum,
                                   float* __restrict__ out_hc) {
  __shared__ float buf[DIM];
  int f = threadIdx.x;  // 128 threads
  buf[f] = colsum[f];
  __syncthreads();
  for (int s = DIM / 2; s > 0; s >>= 1) {
    if (f < s) buf[f] += buf[f + s];
    __syncthreads();
  }
  out_hc[f] = colsum[f] / buf[0];
}

// ------------------------------------------------------- edge scatter (agg)

// One wave32 per edge; each lane moves 4 contiguous floats (float4 gather,
// 4 native f32 atomics into the L2-resident agg buffer).
__global__ void aggregate_kernel(const float* __restrict__ h,
                                 const int* __restrict__ src,
                                 const int* __restrict__ dst,
                                 float* __restrict__ agg) {
  long tid = (long)blockIdx.x * blockDim.x + threadIdx.x;
  int  lane = (int)(tid & 31);
  long e = tid >> 5;
  if (e >= NE) return;
  int s = src[e], d = dst[e];
  float4 v = ((const float4*)(h + (long)s * DIM))[lane];
  float* ap = agg + (long)d * DIM + lane * 4;
  unsafeAtomicAdd(ap + 0, v.x);
  unsafeAtomicAdd(ap + 1, v.y);
  unsafeAtomicAdd(ap + 2, v.z);
  unsafeAtomicAdd(ap + 3, v.w);
}

// -------------------------------------- fused GraphConv GEMM (WMMA f32 path)
// out = relu(agg @ W_rel + h @ W_root + b), one 16x128 output strip per wave.
// Weights staged transposed in LDS: sW[col * WSTRIDE + k], so each
// V_WMMA_F32_16X16X4_F32 B-fragment is one conflict-free ds_load_b64.
// Fragment maps (wave32):
//   A: a[v]   = A[lane%16][k0 + v + 2*(lane>>4)]            (v2f, global b64)
//   B: b[v]   = B[k0 + v + 2*(lane>>4)][col0 + lane%16]     (v2f, LDS b64)
//   C/D: c[j] = D[row0 + j + 8*(lane>>4)][col0 + lane%16]

__global__ void __launch_bounds__(256)
graphconv_gemm_kernel(const float* __restrict__ agg, const float* __restrict__ h,
                      const float* __restrict__ W_rel, const float* __restrict__ W_root,
                      const float* __restrict__ bias, float* __restrict__ out) {
  extern __shared__ float smem[];
  float* sWrel  = smem;                      // [DIM][WSTRIDE] transposed: [col][k]
  float* sWroot = smem + DIM * WSTRIDE;
  float* sBias  = smem + 2 * DIM * WSTRIDE;  // [DIM]

  // Cooperative staging: coalesced global reads, transposed LDS writes.
  for (int idx = threadIdx.x; idx < DIM * DIM; idx += blockDim.x) {
    int k = idx >> 7, col = idx & (DIM - 1);
    sWrel [col * WSTRIDE + k] = W_rel [idx];
    sWroot[col * WSTRIDE + k] = W_root[idx];
  }
  if (threadIdx.x < DIM) sBias[threadIdx.x] = bias[threadIdx.x];
  __syncthreads();

  int wave = blockIdx.x * (blockDim.x >> 5) + (threadIdx.x >> 5);
  int row0 = wave * 16;
  if (row0 >= NN) return;          // wave-uniform: EXEC all-1s at every WMMA
  int lane  = threadIdx.x & 31;
  int m     = lane & 15;
  int khalf = lane >> 4;

  v8f acc[8] = {};                 // 8 col-tiles x 8 VGPRs accumulators

  const float* arow = agg + (long)(row0 + m) * DIM;
  const float* hrow = h   + (long)(row0 + m) * DIM;

  for (int k0 = 0; k0 < DIM; k0 += 4) {
    int ka = k0 + 2 * khalf;
    v2f a_rel  = *(const v2f*)(arow + ka);
    v2f a_root = *(const v2f*)(hrow + ka);
#pragma unroll
    for (int t = 0; t < 8; ++t) {
      int col = t * 16 + m;
      v2f b_rel  = *(const v2f*)(&sWrel [col * WSTRIDE + ka]);
      v2f b_root = *(const v2f*)(&sWroot[col * WSTRIDE + ka]);
      acc[t] = __builtin_amdgcn_wmma_f32_16x16x4_f32(
          false, a_rel, false, b_rel, (short)0, acc[t], false, false);
      acc[t] = __builtin_amdgcn_wmma_f32_16x16x4_f32(
          false, a_root, false, b_root, (short)0, acc[t], false, false);
    }
  }

#pragma unroll
  for (int t = 0; t < 8; ++t) {
#pragma unroll
    for (int j = 0; j < 8; ++j) {
      int row = row0 + j + 8 * khalf;
      int col = t * 16 + m;
      float v = acc[t][j] + sBias[col];
      out[(long)row * DIM + col] = fmaxf(v, 0.0f);
    }
  }
}

// ----------------------------------------------------------- global pooling

__global__ void pool_sum_kernel(const float* __restrict__ h, const int* __restrict__ batch,
                                float* __restrict__ psum, float* __restrict__ pcnt) {
  long tid = (long)blockIdx.x * blockDim.x + threadIdx.x;
  int  lane = (int)(tid & 31);
  long node = tid >> 5;
  if (node >= NN) return;
  int g = batch[node];
  float4 v = ((const float4*)(h + node * DIM))[lane];
  float* sp = psum + (long)g * DIM + lane * 4;
  unsafeAtomicAdd(sp + 0, v.x);
  unsafeAtomicAdd(sp + 1, v.y);
  unsafeAtomicAdd(sp + 2, v.z);
  unsafeAtomicAdd(sp + 3, v.w);
  if (lane == 0) unsafeAtomicAdd(&pcnt[g], 1.0f);
}

__global__ void pool_div_kernel(const float* __restrict__ psum, const float* __restrict__ pcnt,
                                float* __restrict__ out) {
  int i = blockIdx.x * blockDim.x + threadIdx.x;
  if (i >= NG * DIM) return;
  float c = pcnt[i / DIM];
  out[i] = psum[i] / fmaxf(c, 1.0f);
}

// ---------------------------------------------------------------- launcher

extern "C" void kernel_launch(void* const* d_in, const int* in_sizes, int n_in,
                              void* d_out, int out_size, void* d_ws, size_t ws_size,
                              hipStream_t stream) {
  const float* x       = (const float*)d_in[0];
  const int*   ei      = (const int*)  d_in[1];
  const int*   batch   = (const int*)  d_in[2];
  const float* W_rel1  = (const float*)d_in[3];
  const float* W_root1 = (const float*)d_in[4];
  const float* b1      = (const float*)d_in[5];
  const float* W_rel2  = (const float*)d_in[6];
  const float* W_root2 = (const float*)d_in[7];
  const float* b2      = (const float*)d_in[8];
  const float* W_rel3  = (const float*)d_in[9];
  const float* W_root3 = (const float*)d_in[10];
  const float* b3      = (const float*)d_in[11];
  const int* src = ei;          // edge_index[0, :]
  const int* dst = ei + NE;     // edge_index[1, :]

  // Workspace carve-out (~77 MB total), 256B aligned slices.
  char*  ws  = (char*)d_ws;
  size_t off = 0;
  auto carve = [&](size_t bytes) -> float* {
    float* p = (float*)(ws + off);
    off += (bytes + 255) & ~(size_t)255;
    return p;
  };
  float* agg    = carve((size_t)NN * DIM * sizeof(float));
  float* h0     = carve((size_t)NN * DIM * sizeof(float));
  float* h1     = carve((size_t)NN * DIM * sizeof(float));
  float* colsum = carve(DIM * sizeof(float));
  float* psum   = carve((size_t)NG * DIM * sizeof(float));
  float* pcnt   = carve(NG * sizeof(float));

  float* out_pooled = (float*)d_out;            // [64, 128]
  float* out_hc     = out_pooled + NG * DIM;    // [128]

  const long aggN  = (long)NN * DIM;
  const int  zgrid = (int)((aggN + 255) / 256);
  const int  agrid = (int)(((long)NE * 32 + 255) / 256);   // one wave per edge
  const int  ggrid = ((NN + 15) / 16 + 7) / 8;             // 8 waves per block
  const int  pgrid = (int)(((long)NN * 32 + 255) / 256);

  // handcrafted label percentages
  zero_kernel<<<1, 128, 0, stream>>>(colsum, DIM);
  colsum_kernel<<<512, 128, 0, stream>>>(x, colsum);
  handcrafted_kernel<<<1, 128, 0, stream>>>(colsum, out_hc);

  // layer 1: x -> h0
  zero_kernel<<<zgrid, 256, 0, stream>>>(agg, aggN);
  aggregate_kernel<<<agrid, 256, 0, stream>>>(x, src, dst, agg);
  graphconv_gemm_kernel<<<ggrid, 256, SMEM_BYTES, stream>>>(agg, x, W_rel1, W_root1, b1, h0);

  // layer 2: h0 -> h1
  zero_kernel<<<zgrid, 256, 0, stream>>>(agg, aggN);
  aggregate_kernel<<<agrid, 256, 0, stream>>>(h0, src, dst, agg);
  graphconv_gemm_kernel<<<ggrid, 256, SMEM_BYTES, stream>>>(agg, h0, W_rel2, W_root2, b2, h1);

  // layer 3: h1 -> h0
  zero_kernel<<<zgrid, 256, 0, stream>>>(agg, aggN);
  aggregate_kernel<<<agrid, 256, 0, stream>>>(h1, src, dst, agg);
  graphconv_gemm_kernel<<<ggrid, 256, SMEM_BYTES, stream>>>(agg, h1, W_rel3, W_root3, b3, h0);

  // global mean pool
  zero_kernel<<<(NG * DIM + 255) / 256, 256, 0, stream>>>(psum, NG * DIM);
  zero_kernel<<<1, 64, 0, stream>>>(pcnt, NG);
  pool_sum_kernel<<<pgrid, 256, 0, stream>>>(h0, batch, psum, pcnt);
  pool_div_kernel<<<(NG * DIM + 255) / 256, 256, 0, stream>>>(psum, pcnt, out_pooled);
}